// ObjectDetectionModelV2_83760452206679
// MI455X (gfx1250) — compile-verified
//
#include <hip/hip_runtime.h>
#include <hip/hip_bf16.h>

#define NUM_OBJ 32
#define C_IN    256
#define C_MID   128
#define PIX     65536   // 256*256
#define B_BATCH 4

typedef __attribute__((ext_vector_type(2))) float v2f;
typedef __attribute__((ext_vector_type(8))) float v8f;
typedef __attribute__((ext_vector_type(4))) float v4f;
typedef __attribute__((ext_vector_type(4))) int   v4i;

// ---------------------------------------------------------------------------
// Kernel 0: zero the pooled scratch (B*32*256 floats) every call.
// ---------------------------------------------------------------------------
__global__ void zero_pooled_k(float* __restrict__ pooled) {
    pooled[blockIdx.x * 256 + threadIdx.x] = 0.0f;
}

// ---------------------------------------------------------------------------
// Kernel 1: segmented max over pixels, per (batch, channel).
// grid = (256 channels, 4 batches), block = 256 threads (8 wave32).
// smax[id*256 + tid]: per-thread private column -> no races, no bank conflicts
// in the hot loop (stride 256 => bank = tid mod 64).
// Seeded with 0.0f => implements reference's max(seg, 0) clamp + empty fill.
// encoded is a 256 MiB single-use stream -> non-temporal loads (TH_NT) so it
// does not evict the L2-resident ids array (re-read 256x per batch).
// ---------------------------------------------------------------------------
__global__ __launch_bounds__(256) void segmax_k(const float* __restrict__ encoded,
                                                const int*   __restrict__ ids,
                                                float*       __restrict__ pooled) {
    __shared__ float smax[33 * 256];
    __shared__ float part[33 * 8];
    const int tid = threadIdx.x;
    const int c   = blockIdx.x;
    const int b   = blockIdx.y;

    #pragma unroll 4
    for (int i = tid; i < 33 * 256; i += 256) smax[i] = 0.0f;
    __syncthreads();

    const v4f* __restrict__ e4 = (const v4f*)(encoded + ((size_t)b * C_IN + c) * PIX);
    const v4i* __restrict__ i4 = (const v4i*)(ids + (size_t)b * PIX);

    // 64 iterations: each wave reads 512B contiguous encoded + 512B ids (L2-hot)
    for (int it = 0; it < PIX / (256 * 4); ++it) {
        v4f e  = __builtin_nontemporal_load(&e4[it * 256 + tid]);
        v4i id = i4[it * 256 + tid];
        int a0 = id.x * 256 + tid; if (e.x > smax[a0]) smax[a0] = e.x;
        int a1 = id.y * 256 + tid; if (e.y > smax[a1]) smax[a1] = e.y;
        int a2 = id.z * 256 + tid; if (e.z > smax[a2]) smax[a2] = e.z;
        int a3 = id.w * 256 + tid; if (e.w > smax[a3]) smax[a3] = e.w;
    }
    __syncthreads();

    // Stage 1: 256 threads = 32 ids (1..32) x 8 parts; id-staggered column
    // order gives 4-way banking (optimal for 256 lanes / 64 banks).
    {
        int id = 1 + (tid & 31);
        int p  = tid >> 5;
        int base = id * 256 + p * 32;
        float m = 0.0f;
        #pragma unroll
        for (int jj = 0; jj < 32; ++jj) {
            int j = (jj + id) & 31;
            m = fmaxf(m, smax[base + j]);
        }
        part[id * 8 + p] = m;
    }
    __syncthreads();

    // Stage 2: threads 1..32 fold 8 partials and publish via int-ordered
    // atomic max (all values >= 0 so float order == int order).
    if (tid >= 1 && tid <= 32) {
        float m = 0.0f;
        #pragma unroll
        for (int p = 0; p < 8; ++p) m = fmaxf(m, part[tid * 8 + p]);
        atomicMax((int*)(pooled + ((size_t)b * NUM_OBJ + (tid - 1)) * C_IN + c),
                  __float_as_int(m));
    }
}

// ---------------------------------------------------------------------------
// Kernel 2: h = pooled[128,256] @ w1[256,128] + b1, via V_WMMA_F32_16X16X4_F32.
// One wave per 16x16 output tile; grid = (8 tilesN, 8 tilesM).
// A layout (16x4 f32): lane l -> row = l&15, k = kbase + 2*(l>>4) + {0,1}
// B layout (4x16 f32): lane l -> col = l&15, k = kbase + 2*(l>>4) + {0,1}
// D layout (16x16 f32, 8 VGPRs): vg -> row = 8*(l>>4) + vg, col = l&15
// ---------------------------------------------------------------------------
__global__ __launch_bounds__(32) void gemm1_k(const float* __restrict__ pooled,
                                              const float* __restrict__ w1,
                                              const float* __restrict__ b1,
                                              float*       __restrict__ h) {
    const int lane = threadIdx.x;
    const int half = lane >> 4;      // 0 or 1
    const int l15  = lane & 15;
    const int row  = blockIdx.y * 16 + l15;   // A row (M)
    const int col  = blockIdx.x * 16 + l15;   // B col (N)

    v8f acc = {};
    #pragma unroll 4
    for (int k0 = 0; k0 < C_IN; k0 += 4) {
        const int k = k0 + half * 2;
        v2f a; a.x = pooled[row * C_IN + k];
               a.y = pooled[row * C_IN + k + 1];
        v2f bb; bb.x = w1[(size_t)k * C_MID + col];
                bb.y = w1[(size_t)(k + 1) * C_MID + col];
        acc = __builtin_amdgcn_wmma_f32_16x16x4_f32(
            /*neg_a=*/false, a, /*neg_b=*/false, bb,
            /*c_mod=*/(short)0, acc, /*reuse_a=*/false, /*reuse_b=*/false);
    }

    const float bias = b1[col];
    #pragma unroll
    for (int vg = 0; vg < 8; ++vg) {
        const int m = blockIdx.y * 16 + half * 8 + vg;
        h[m * C_MID + col] = acc[vg] + bias;
    }
}

// ---------------------------------------------------------------------------
// Kernel 3: bbox = sigmoid(h[128,128] @ w2[128,4] + b2) -> 512 floats.
// ---------------------------------------------------------------------------
__global__ __launch_bounds__(128) void head_k(const float* __restrict__ h,
                                              const float* __restrict__ w2,
                                              const float* __restrict__ b2,
                                              float*       __restrict__ out) {
    const int m = threadIdx.x;   // 0..127 = b*32 + n
    float acc0 = b2[0], acc1 = b2[1], acc2 = b2[2], acc3 = b2[3];
    #pragma unroll 8
    for (int k = 0; k < C_MID; ++k) {
        const float hv = h[m * C_MID + k];
        acc0 += hv * w2[k * 4 + 0];
        acc1 += hv * w2[k * 4 + 1];
        acc2 += hv * w2[k * 4 + 2];
        acc3 += hv * w2[k * 4 + 3];
    }
    out[m * 4 + 0] = 1.0f / (1.0f + __expf(-acc0));
    out[m * 4 + 1] = 1.0f / (1.0f + __expf(-acc1));
    out[m * 4 + 2] = 1.0f / (1.0f + __expf(-acc2));
    out[m * 4 + 3] = 1.0f / (1.0f + __expf(-acc3));
}

// ---------------------------------------------------------------------------
extern "C" void kernel_launch(void* const* d_in, const int* in_sizes, int n_in,
                              void* d_out, int out_size, void* d_ws, size_t ws_size,
                              hipStream_t stream) {
    const float* encoded = (const float*)d_in[0];   // [4,256,256,256]
    const float* w1      = (const float*)d_in[1];   // [256,128]
    const float* b1      = (const float*)d_in[2];   // [128]
    const float* w2      = (const float*)d_in[3];   // [128,4]
    const float* b2      = (const float*)d_in[4];   // [4]
    const int*   masks   = (const int*)d_in[5];     // [4,1,256,256]
    float*       out     = (float*)d_out;           // [4,32,4]

    float* pooled = (float*)d_ws;                          // 4*32*256 floats
    float* h      = pooled + (size_t)B_BATCH * NUM_OBJ * C_IN;  // 128*128 floats

    zero_pooled_k<<<(B_BATCH * NUM_OBJ * C_IN) / 256, 256, 0, stream>>>(pooled);
    segmax_k<<<dim3(C_IN, B_BATCH), 256, 0, stream>>>(encoded, masks, pooled);
    gemm1_k<<<dim3(C_MID / 16, (B_BATCH * NUM_OBJ) / 16), 32, 0, stream>>>(pooled, w1, b1, h);
    head_k<<<1, 128, 0, stream>>>(h, w2, b2, out);
}